// Colorizer_61186104099689
// MI455X (gfx1250) — compile-verified
//
#include <hip/hip_runtime.h>
#include <stdint.h>

typedef float v2f __attribute__((ext_vector_type(2)));
typedef float v8f __attribute__((ext_vector_type(8)));

#define H1 128
#define W1 128
#define CF 256
#define NB 4
#define PS 13          // patch size
#define RR 6           // radius
#define NC 16          // classes
#define CH_STRIDE (H1*W1)   // 16384 floats between channels

// ---------------------------------------------------------------------------
// Kernel 1: bilinear 1/4 downsample of int labels at half-pixel centers.
// Sample point = (4y+1.5, 4x+1.5) -> average of 4 pixels, truncate = sum>>2.
// ---------------------------------------------------------------------------
__global__ void prep_labels_kernel(const int* __restrict__ q, uint8_t* __restrict__ lab) {
    int i = blockIdx.x * blockDim.x + threadIdx.x;        // over 4*128*128
    if (i >= NB * H1 * W1) return;
    int x = i & (W1 - 1);
    int y = (i >> 7) & (H1 - 1);
    int b = i >> 14;
    const int* qb = q + (size_t)b * 512 * 512;
    int r0 = 4 * y + 1, c0 = 4 * x + 1;
    int s = qb[r0 * 512 + c0] + qb[r0 * 512 + c0 + 1]
          + qb[(r0 + 1) * 512 + c0] + qb[(r0 + 1) * 512 + c0 + 1];
    lab[i] = (uint8_t)(s >> 2);
}

// ---------------------------------------------------------------------------
// Kernel 2: correlation via V_WMMA_F32_16X16X4_F32, masked softmax over 169
// taps, class-binned weighted sum. One block = (b, y, 16-wide x tile).
// 4 waves/block; wave w owns dy = w, w+4, w+8(,12).
// ---------------------------------------------------------------------------
__global__ __launch_bounds__(128)
void colorizer_main_kernel(const float* __restrict__ fr, const float* __restrict__ ft,
                           const uint8_t* __restrict__ lab, float* __restrict__ out) {
    __shared__ float corrS[(PS * PS + 1) * 16];  // [169][16 x] + dummy row 169
    __shared__ float Pscr[4][16 * 32];           // per-wave 16x32 product scratch
    __shared__ int   labS[PS * 28];              // clamped label tile [13][28]
    __shared__ float redMax[128];
    __shared__ float redSum[128];

    int bid = blockIdx.x;                   // 0..4095
    int x0  = (bid & 7) * 16;
    int y   = (bid >> 3) & (H1 - 1);
    int b   = bid >> 10;

    int tid  = threadIdx.x;
    int lane = tid & 31;
    int w    = tid >> 5;

    // preload edge-clamped labels: rows y-6..y+6, cols x0-6..x0+21
    const uint8_t* labB = lab + (size_t)b * H1 * W1;
    for (int e = tid; e < PS * 28; e += 128) {
        int rr = e / 28, cc = e % 28;
        int gy = min(max(y + rr - RR, 0), H1 - 1);
        int gx = min(max(x0 + cc - RR, 0), W1 - 1);
        labS[e] = (int)labB[gy * W1 + gx];
    }

    // WMMA fragment lane mapping (f32 16x16x4): lane = {half, idx}
    int half = lane >> 4;
    int idx  = lane & 15;

    // A: feats_t[b, k, y, x0+idx]; channel k = k0 + 2*half + vreg
    const float* ftB = ft + (size_t)b * CF * CH_STRIDE
                          + (size_t)(2 * half) * CH_STRIDE
                          + (size_t)y * W1 + (x0 + idx);
    // B tiles: feats_r columns x0-6+idx (tile0), +16 (tile1); edge-clamped.
    int bc0 = min(max(x0 - RR + idx, 0), W1 - 1);
    int bc1 = min(max(x0 - RR + 16 + idx, 0), W1 - 1);
    const float* frB = fr + (size_t)b * CF * CH_STRIDE + (size_t)(2 * half) * CH_STRIDE;

    for (int dy = w; dy < PS; dy += 4) {
        v8f acc0 = {0.f,0.f,0.f,0.f,0.f,0.f,0.f,0.f};
        v8f acc1 = {0.f,0.f,0.f,0.f,0.f,0.f,0.f,0.f};
        int ry = y + dy - RR;
        if (ry >= 0 && ry < H1) {                   // wave-uniform branch
            const float* ap  = ftB;
            const float* bp0 = frB + (size_t)ry * W1 + bc0;
            const float* bp1 = frB + (size_t)ry * W1 + bc1;
            #pragma unroll 4
            for (int k = 0; k < CF / 4; ++k) {
                v2f a, b0, b1;
                a.x  = ap[0];
                a.y  = ap[CH_STRIDE];
                b0.x = bp0[0];
                b0.y = bp0[CH_STRIDE];
                b1.x = bp1[0];
                b1.y = bp1[CH_STRIDE];
                acc0 = __builtin_amdgcn_wmma_f32_16x16x4_f32(
                    false, a, false, b0, (short)0, acc0, false, false);
                acc1 = __builtin_amdgcn_wmma_f32_16x16x4_f32(
                    false, a, false, b1, (short)0, acc1, false, false);
                ap  += 4 * CH_STRIDE;
                bp0 += 4 * CH_STRIDE;
                bp1 += 4 * CH_STRIDE;
            }
        }
        // dump 16x32 product to per-wave LDS scratch
        float* Pw = Pscr[w];
        #pragma unroll
        for (int v = 0; v < 8; ++v) {
            int m = v + 8 * half;
            Pw[m * 32 + idx]      = acc0[v];
            Pw[m * 32 + idx + 16] = acc1[v];
        }
        __builtin_amdgcn_wave_barrier();   // order intra-wave LDS store -> cross-lane read
        // extract band corr[m,dx] = P[m, m+dx]; 7 uniform steps, no branches.
        // Out-of-range dx -> dummy row 169; out-of-range column -> 0 (zero-pad).
        #pragma unroll
        for (int i = 0; i < 7; ++i) {
            int e  = lane + 32 * i;        // 0..223
            int dx = e >> 4;               // 0..13
            int m  = e & 15;
            int dxc   = (dx < PS) ? dx : (PS - 1);
            float v   = Pw[m * 32 + m + dxc];
            int gcol  = x0 + m + dxc - RR;
            v = (gcol >= 0 && gcol < W1) ? v : 0.0f;
            int p = (dx < PS) ? (dy * PS + dx) : (PS * PS);
            corrS[p * 16 + m] = v;
        }
        __builtin_amdgcn_wave_barrier();
    }
    __syncthreads();

    // ---- masked softmax over 169 taps, parallel over (x, group g of 8) ----
    int x = tid & 15;
    int g = tid >> 4;                       // 0..7, handles p = g, g+8, ...
    float mx = -3.0e38f;
    for (int p = g; p < PS * PS; p += 8) {
        float c = corrS[p * 16 + x];
        float v = (c == 0.0f) ? -1.0e10f : c;
        mx = fmaxf(mx, v);
    }
    redMax[tid] = mx;
    __syncthreads();
    mx = redMax[x];
    #pragma unroll
    for (int j = 1; j < 8; ++j) mx = fmaxf(mx, redMax[j * 16 + x]);

    float s = 0.0f;
    for (int p = g; p < PS * PS; p += 8) {
        float c = corrS[p * 16 + x];
        float v = (c == 0.0f) ? -1.0e10f : c;
        float e = __expf(v - mx);
        corrS[p * 16 + x] = e;
        s += e;
    }
    redSum[tid] = s;
    __syncthreads();                        // also publishes exp'd corrS
    float sum = redSum[x];
    #pragma unroll
    for (int j = 1; j < 8; ++j) sum += redSum[j * 16 + x];
    float inv = 1.0f / sum;

    // ---- each thread owns classes 2g, 2g+1 at column x ----
    int c0 = 2 * g, c1 = c0 + 1;
    float a0 = 0.0f, a1 = 0.0f;
    for (int dy = 0; dy < PS; ++dy) {
        #pragma unroll
        for (int dx = 0; dx < PS; ++dx) {
            float wv = corrS[(dy * PS + dx) * 16 + x];
            int  cl  = labS[dy * 28 + x + dx];
            a0 += (cl == c0) ? wv : 0.0f;
            a1 += (cl == c1) ? wv : 0.0f;
        }
    }
    a0 *= inv;
    a1 *= inv;

    // write out[b, c, y, x0+x]
    float* outB = out + (size_t)b * NC * CH_STRIDE + (size_t)y * W1 + x0 + x;
    outB[(size_t)c0 * CH_STRIDE] = a0;
    outB[(size_t)c1 * CH_STRIDE] = a1;
}

// ---------------------------------------------------------------------------
extern "C" void kernel_launch(void* const* d_in, const int* in_sizes, int n_in,
                              void* d_out, int out_size, void* d_ws, size_t ws_size,
                              hipStream_t stream) {
    const float* feats_r = (const float*)d_in[0];
    const float* feats_t = (const float*)d_in[1];
    const int*   quant_r = (const int*)d_in[2];
    float* out = (float*)d_out;
    uint8_t* lab = (uint8_t*)d_ws;   // 4*128*128 = 64 KB

    int nlab = NB * H1 * W1;
    prep_labels_kernel<<<(nlab + 255) / 256, 256, 0, stream>>>(quant_r, lab);

    int nblocks = NB * H1 * (W1 / 16);   // 4096
    colorizer_main_kernel<<<nblocks, 128, 0, stream>>>(feats_r, feats_t, lab, out);
}